// FractalBrownianMotion_46351287058971
// MI455X (gfx1250) — compile-verified
//
#include <hip/hip_runtime.h>
#include <cstdint>

typedef float v4f __attribute__((ext_vector_type(4)));

#define HW      1024
#define NPIX    (HW * HW)        // 1048576 pixels in noise plane
#define PLANE4  (NPIX / 4)       // 262144 float4 per image plane
#define NPLANES 48               // 16 batch * 3 channels
#define PPB     4                // planes per block in apply kernel
#define NOCT    5
#define NTAB    5709             // total (cos,sin) table entries over 5 octaves

// ---------- float <-> order-preserving uint (for atomic min/max) ----------
__device__ __forceinline__ unsigned f2ord(float f) {
    unsigned u = __float_as_uint(f);
    return (u & 0x80000000u) ? ~u : (u ^ 0x80000000u);
}
__device__ __forceinline__ float ord2f(unsigned u) {
    return __uint_as_float((u & 0x80000000u) ? (u ^ 0x80000000u) : ~u);
}

// ---------- Perlin octave value from precomputed (cos,sin) table ----------
// tab points at this octave's table; entry k = (cos(angle_k), sin(angle_k)),
// angles grid is (ry+1) x (rx+1), row-major.
__device__ __forceinline__ float perlin_val(const float2* __restrict__ tab,
                                            int i, int j, int o) {
    const int   ry     = 4 << o;              // == rx
    const int   shift  = 8 - o;               // cell size = 1024/ry = 1<<shift
    const float step   = (float)ry * (1.0f / 1023.0f);  // linspace(0,ry,1024)
    float y  = (float)i * step;
    float x  = (float)j * step;
    float P0 = y - floorf(y);
    float P1 = x - floorf(x);
    int gy = i >> shift;
    int gx = j >> shift;
    int stride = ry + 1;
    float2 g00 = tab[gy * stride + gx];
    float2 g01 = tab[gy * stride + gx + 1];
    float2 g10 = tab[(gy + 1) * stride + gx];
    float2 g11 = tab[(gy + 1) * stride + gx + 1];
    float n00 = P0 * g00.x + P1 * g00.y;
    float n10 = (P0 - 1.0f) * g10.x + P1 * g10.y;
    float n01 = P0 * g01.x + (P1 - 1.0f) * g01.y;
    float n11 = (P0 - 1.0f) * g11.x + (P1 - 1.0f) * g11.y;
    float wy = P0 * P0 * (3.0f - 2.0f * P0);
    float wx = P1 * P1 * (3.0f - 2.0f * P1);
    float n0 = n00 + wy * (n10 - n00);
    float n1 = n01 + wy * (n11 - n01);
    return n0 + wx * (n1 - n0);
}

// ---------- kernel 0: merged prep: reset min/max slots + build trig tables ----------
__global__ void __launch_bounds__(256)
fbm_prep(const float* __restrict__ a0, const float* __restrict__ a1,
         const float* __restrict__ a2, const float* __restrict__ a3,
         const float* __restrict__ a4,
         float2* __restrict__ tab, unsigned* __restrict__ mm) {
    int g = blockIdx.x * 256 + threadIdx.x;
    if (g < 2 * NOCT) {
        // even slot = min (init +inf-ord), odd slot = max (init 0)
        mm[g] = (g & 1) ? 0u : 0xFFFFFFFFu;
    }
    if (g < NTAB) {
        const float* a;
        int base;
        if      (g < 25)   { a = a0; base = 0;    }
        else if (g < 106)  { a = a1; base = 25;   }
        else if (g < 395)  { a = a2; base = 106;  }
        else if (g < 1484) { a = a3; base = 395;  }
        else               { a = a4; base = 1484; }
        float ang = a[g - base];
        tab[g] = make_float2(cosf(ang), sinf(ang));
    }
}

// ---------- kernel 1: per-octave global min/max over the 1024x1024 field ----------
__global__ void __launch_bounds__(256)
fbm_minmax(const float2* __restrict__ tab, unsigned* __restrict__ mm) {
    __shared__ unsigned smn[NOCT], smx[NOCT];
    int t = threadIdx.x;
    if (t < NOCT) { smn[t] = 0xFFFFFFFFu; smx[t] = 0u; }
    __syncthreads();
    int idx = blockIdx.x * 256 + t;
    int i = idx >> 10;
    int j = idx & (HW - 1);
    const int tbase[NOCT] = {0, 25, 106, 395, 1484};
    #pragma unroll
    for (int o = 0; o < NOCT; ++o) {
        float n = perlin_val(tab + tbase[o], i, j, o);
        unsigned u = f2ord(n);
        unsigned umn = u, umx = u;
        #pragma unroll
        for (int off = 16; off > 0; off >>= 1) {   // wave32 butterfly
            unsigned omn = (unsigned)__shfl_xor((int)umn, off, 32);
            unsigned omx = (unsigned)__shfl_xor((int)umx, off, 32);
            umn = (omn < umn) ? omn : umn;
            umx = (omx > umx) ? omx : umx;
        }
        if ((t & 31) == 0) {
            atomicMin(&smn[o], umn);
            atomicMax(&smx[o], umx);
        }
    }
    __syncthreads();
    if (t < NOCT) {
        atomicMin(&mm[2 * t], smn[t]);
        atomicMax(&mm[2 * t + 1], smx[t]);
    }
}

// ---------- kernel 2: combined normalized noise plane ----------
__global__ void __launch_bounds__(256)
fbm_combine(const float2* __restrict__ tab, const unsigned* __restrict__ mm,
            float* __restrict__ noise) {
    int idx = blockIdx.x * 256 + threadIdx.x;
    int i = idx >> 10;
    int j = idx & (HW - 1);
    const int tbase[NOCT] = {0, 25, 106, 395, 1484};
    float acc = 0.0f;
    float amp = 0.1f;                 // AMPLITUDE * PERSISTENCE^o
    #pragma unroll
    for (int o = 0; o < NOCT; ++o) {
        float n  = perlin_val(tab + tbase[o], i, j, o);
        float mn = ord2f(mm[2 * o]);
        float mx = ord2f(mm[2 * o + 1]);
        float nn = (n - mn) / (mx - mn);
        acc += (nn * 2.0f - 1.0f) * amp;
        amp *= 0.5f;
    }
    noise[idx] = acc;
}

// ---------- kernel 3: bandwidth kernel: out = clip(img + noise*mix, 0, 1) ----------
// Each block owns a 1024-float chunk of the noise plane: async-copied into LDS
// once (CDNA5 GLOBAL_LOAD_ASYNC_TO_LDS + s_wait_asynccnt), reused for PPB planes.
__global__ void __launch_bounds__(256)
fbm_apply(const v4f* __restrict__ img, const float* __restrict__ noise,
          const float* __restrict__ mixp, v4f* __restrict__ out) {
    __shared__ v4f lno[256];
    int t  = threadIdx.x;
    int c  = blockIdx.x;          // chunk: 0..1023 (256 float4 each)
    int pg = blockIdx.y;          // plane group: 0..NPLANES/PPB-1

    const v4f* gsrc = ((const v4f*)noise) + (c * 256 + t);
    unsigned ldsoff = (unsigned)(uintptr_t)(&lno[t]);   // low 32 bits = LDS byte offset
    asm volatile("global_load_async_to_lds_b128 %0, %1, off"
                 :: "v"(ldsoff), "v"(gsrc) : "memory");

    float mix = *mixp;            // uniform scalar load, overlaps async copy

    asm volatile("s_wait_asynccnt 0" ::: "memory");
    __syncthreads();

    v4f nm = lno[t] * mix;        // ds_load_b128
    v4f zero = 0.0f, one = 1.0f;
    int base4 = c * 256 + t;
    #pragma unroll
    for (int p = 0; p < PPB; ++p) {
        int plane = pg * PPB + p;
        size_t e = (size_t)plane * PLANE4 + base4;
        if (p + 1 < PPB)
            __builtin_prefetch((const void*)(img + e + PLANE4), 0, 0);  // global_prefetch_b8
        v4f v = __builtin_nontemporal_load(img + e);                    // streamed once
        v = __builtin_elementwise_min(__builtin_elementwise_max(v + nm, zero), one);
        __builtin_nontemporal_store(v, out + e);                        // streamed once
    }
}

// ---------- host-side orchestration (4 launches total) ----------
extern "C" void kernel_launch(void* const* d_in, const int* in_sizes, int n_in,
                              void* d_out, int out_size, void* d_ws, size_t ws_size,
                              hipStream_t stream) {
    const float* img = (const float*)d_in[0];
    const float* mix = (const float*)d_in[1];

    // workspace layout: [0,40) min/max uints; +256 trig table; +64KiB noise plane
    unsigned* mm    = (unsigned*)d_ws;
    float2*   tab   = (float2*)((char*)d_ws + 256);
    float*    noise = (float*)((char*)d_ws + 65536);

    fbm_prep<<<(NTAB + 255) / 256, 256, 0, stream>>>(
        (const float*)d_in[2], (const float*)d_in[3], (const float*)d_in[4],
        (const float*)d_in[5], (const float*)d_in[6], tab, mm);

    fbm_minmax<<<NPIX / 256, 256, 0, stream>>>(tab, mm);
    fbm_combine<<<NPIX / 256, 256, 0, stream>>>(tab, mm, noise);

    dim3 grid(HW, NPLANES / PPB);   // 1024 chunks x 12 plane-groups
    fbm_apply<<<grid, 256, 0, stream>>>((const v4f*)img, noise, mix, (v4f*)d_out);
}